// ZINBDecoder_76184129896495
// MI455X (gfx1250) — compile-verified
//
#include <hip/hip_runtime.h>
#include <math.h>

typedef float v2f __attribute__((ext_vector_type(2)));
typedef float v8f __attribute__((ext_vector_type(8)));

#define WAVE            32
#define D               128
#define TILE            16      // edges per wave-tile (rows of the WMMA A/D matrices)
#define WAVES_PER_BLOCK 8
#define BLOCK           (WAVE * WAVES_PER_BLOCK)
#define KFRAGS          (D / 4) // 32 chained V_WMMA_F32_16X16X4_F32 per tile

__device__ __forceinline__ float sigmoidf_(float x) {
    return 1.0f / (1.0f + __expf(-x));
}
__device__ __forceinline__ float softplusf_(float x) {
    // numerically-stable log1p(exp(x))
    return (x > 20.0f) ? x : log1pf(__expf(x));
}

__global__ void __launch_bounds__(BLOCK) zinb_decoder_kernel(
    const float* __restrict__ ufeats, const float* __restrict__ ifeats,
    const float* __restrict__ ge_factor, const float* __restrict__ sz_factor,
    const float* __restrict__ W_mean, const float* __restrict__ b_mean,
    const float* __restrict__ W_disp, const float* __restrict__ b_disp,
    const float* __restrict__ W_pi,   const float* __restrict__ b_pi,
    const int* __restrict__ src_idx,  const int* __restrict__ dst_idx,
    float* __restrict__ out, long long E)
{
    const int lane = threadIdx.x & (WAVE - 1);
    const int w    = threadIdx.x >> 5;
    const int n    = lane & 15;   // WMMA column / edge-in-tile
    const int half = lane >> 4;   // K-half of A fragment (per ISA A 16x4 layout)

    const long long nTiles = (E + TILE - 1) / TILE;
    const long long waveId = (long long)blockIdx.x * WAVES_PER_BLOCK + w;
    const long long nWaves = (long long)gridDim.x * WAVES_PER_BLOCK;

    const float bm = b_mean[0], bd = b_disp[0], bp = b_pi[0];

#if __has_builtin(__builtin_amdgcn_wmma_f32_16x16x4_f32)
    __shared__ float xch[WAVES_PER_BLOCK][3][TILE];

    // ---- Build B fragments once per wave: B = [W_mean | W_disp | W_pi | 0..] (128x16)
    // 32-bit B 4x16 layout: VGPR0 holds K=kb+2*half, VGPR1 holds K=kb+2*half+1.
    const float* Wsel = (n == 0) ? W_mean : (n == 1) ? W_disp : W_pi;
    v2f Bf[KFRAGS];
    #pragma unroll
    for (int kk = 0; kk < KFRAGS; ++kk) {
        float b0 = 0.0f, b1 = 0.0f;
        if (n < 3) {
            const int kb = kk * 4 + half * 2;
            b0 = Wsel[kb];
            b1 = Wsel[kb + 1];
        }
        Bf[kk].x = b0; Bf[kk].y = b1;
    }

    for (long long tile = waveId; tile < nTiles; tile += nWaves) {
        const long long base = tile * TILE;
        long long e = base + n;
        if (e >= E) e = E - 1;                 // clamp: keeps EXEC all-ones for WMMA
        const int si = src_idx[e];
        const int di = dst_idx[e];
        const float* up = ufeats + (long long)si * D + half * 2;
        const float* vp = ifeats + (long long)di * D + half * 2;

        v8f acc = {};
        #pragma unroll
        for (int kk = 0; kk < KFRAGS; ++kk) {
            // A 16x4 f32 fragment: this lane supplies h[edge=n, kb+2*half .. +1]
            v2f u2 = *(const v2f*)(up + kk * 4);
            v2f vv = *(const v2f*)(vp + kk * 4);
            v2f a;
            a.x = u2.x * vv.x;
            a.y = u2.y * vv.y;
            // D(16x16,f32) += A(16x4,f32) x B(4x16,f32)
            acc = __builtin_amdgcn_wmma_f32_16x16x4_f32(
                false, a, false, Bf[kk], (short)0, acc, false, false);
        }

        // C/D layout: acc[g] = D[M = g + 8*half][N = n]. Only N=0..2 are real.
        if (n < 3) {
            #pragma unroll
            for (int g = 0; g < 8; ++g)
                xch[w][n][half * 8 + g] = acc[g];
        }
        // wave-private LDS exchange: DS ops are in-order per wave, just drain DScnt
        asm volatile("s_wait_dscnt 0" ::: "memory");

        if (half == 0) {
            const long long eo = base + n;
            if (eo < E) {
                const float dm  = xch[w][0][n];
                const float dd  = xch[w][1][n];
                const float dp  = xch[w][2][n];
                const float gef = ge_factor[di];
                const float szf = sz_factor[si];

                const float mu_   = sigmoidf_(dm + bm);
                const float disp_ = dd + bd;
                const float pi    = sigmoidf_(dp + bp);
                const float disp  = fminf(fmaxf(softplusf_(gef * disp_), 1e-4f), 1e4f);
                const float mu    = szf * fminf(fmaxf(__expf(gef * mu_) - 1.0f, 1e-5f), 1e6f);

                out[eo]             = mu;
                out[E + eo]         = disp;
                out[2 * E + eo]     = pi;
            }
        }
    }
#else
    // Fallback: pure VALU per-lane edge processing (correct, no WMMA)
    const long long tid = waveId * WAVE + lane;
    const long long stride = nWaves * WAVE;
    for (long long e = tid; e < E; e += stride) {
        const int si = src_idx[e];
        const int di = dst_idx[e];
        const float* up = ufeats + (long long)si * D;
        const float* vp = ifeats + (long long)di * D;
        float dm = 0.f, dd = 0.f, dp = 0.f;
        #pragma unroll 8
        for (int k = 0; k < D; ++k) {
            const float h = up[k] * vp[k];
            dm = fmaf(h, W_mean[k], dm);
            dd = fmaf(h, W_disp[k], dd);
            dp = fmaf(h, W_pi[k],   dp);
        }
        const float gef = ge_factor[di];
        const float szf = sz_factor[si];
        const float mu_   = sigmoidf_(dm + bm);
        const float disp_ = dd + bd;
        const float pi    = sigmoidf_(dp + bp);
        const float disp  = fminf(fmaxf(softplusf_(gef * disp_), 1e-4f), 1e4f);
        const float mu    = szf * fminf(fmaxf(__expf(gef * mu_) - 1.0f, 1e-5f), 1e6f);
        out[e]         = mu;
        out[E + e]     = disp;
        out[2 * E + e] = pi;
    }
#endif
}

extern "C" void kernel_launch(void* const* d_in, const int* in_sizes, int n_in,
                              void* d_out, int out_size, void* d_ws, size_t ws_size,
                              hipStream_t stream) {
    const float* ufeats    = (const float*)d_in[0];
    const float* ifeats    = (const float*)d_in[1];
    const float* ge_factor = (const float*)d_in[2];
    const float* sz_factor = (const float*)d_in[3];
    const float* W_mean    = (const float*)d_in[4];
    const float* b_mean    = (const float*)d_in[5];
    const float* W_disp    = (const float*)d_in[6];
    const float* b_disp    = (const float*)d_in[7];
    const float* W_pi      = (const float*)d_in[8];
    const float* b_pi      = (const float*)d_in[9];
    const int*   src_idx   = (const int*)d_in[10];
    const int*   dst_idx   = (const int*)d_in[11];

    const long long E = in_sizes[10];
    const long long nTiles = (E + TILE - 1) / TILE;
    long long blocks = (nTiles + WAVES_PER_BLOCK - 1) / WAVES_PER_BLOCK;
    if (blocks > 2048) blocks = 2048;   // persistent waves; B fragments amortized
    if (blocks < 1)    blocks = 1;

    zinb_decoder_kernel<<<(int)blocks, BLOCK, 0, stream>>>(
        ufeats, ifeats, ge_factor, sz_factor,
        W_mean, b_mean, W_disp, b_disp, W_pi, b_pi,
        src_idx, dst_idx, (float*)d_out, E);
}